// DEQFixedPoint_24592982737220
// MI455X (gfx1250) — compile-verified
//
#include <hip/hip_runtime.h>
#include <hip/hip_bf16.h>

typedef __attribute__((ext_vector_type(2))) float v2f;
typedef __attribute__((ext_vector_type(8))) float v8f;

#define BDIM 256
#define DDIM 2048
#define BD   (BDIM * DDIM)   // 524288
#define TMAX 12
#define NT   256
#define KBLK 64              // K-panel staged in LDS per block iteration

// ---------------------------------------------------------------------------
// GEMM: Out[256,2048] = epilogue( A[256,2048] @ Bm[2048,2048] )
// mode 0: Out = acc + C2[col]                (C2 = bias vector [2048])
// mode 1: Out = tanh(acc + C2[idx]) - A[idx] (g(z) = f(z,x) - z, C2 = xUb)
// mode 2: Out = tanh(acc + C2[idx])          (final f_layer output)
//
// Blocking: 64 blocks = 16 M-tiles x 4 N-blocks. Block = 8 waves; each wave
// owns 4 adjacent 16x16 N-tiles (64 cols) -> one A fragment feeds 4 WMMAs.
// A M-panel (16 x KBLK) staged in LDS via CDNA5 async global->LDS DMA
// (global_load_async_to_lds_b128, ASYNCcnt), padded stride to avoid 16-way
// LDS bank conflicts on same-column row reads.
// ---------------------------------------------------------------------------
__global__ __launch_bounds__(256) void k_gemm_wmma(
        const float* __restrict__ A,
        const float* __restrict__ Bm,
        const float* __restrict__ C2,
        float* __restrict__ Out,
        int mode, const int* __restrict__ act)
{
    if (act && *act == 0) return;

    __shared__ float As[16][KBLK + 4];        // +4 pad: row stride 68 -> 4-bank skew

    const int tid  = threadIdx.x;
    const int lane = tid & 31;
    const int wave = tid >> 5;
    const int mtile = blockIdx.x >> 2;        // 0..15
    const int nblk  = blockIdx.x & 3;         // 0..3
    const int m0  = mtile * 16;
    const int n0w = nblk * 512 + wave * 64;   // this wave's first column
    const int hw  = lane >> 4;                // K offset 0 or 2
    const int l16 = lane & 15;

    // async staging: each of 256 threads moves one 16B chunk of the A panel
    const int arow = tid >> 4;                // 0..15
    const int akc  = (tid & 15) << 2;         // 0,4,...,60
    const unsigned lds_off =
        (unsigned)(size_t)(&As[arow][akc]);   // addr[31:0] == LDS byte offset

    v8f c0 = {0.f,0.f,0.f,0.f,0.f,0.f,0.f,0.f};
    v8f c1 = c0, c2 = c0, c3 = c0;

    for (int k0 = 0; k0 < DDIM; k0 += KBLK) {
        unsigned long long ga =
            (unsigned long long)(size_t)(A + (size_t)(m0 + arow) * DDIM + k0 + akc);
        __syncthreads();                      // LDS consumers from prev iter done
        asm volatile("global_load_async_to_lds_b128 %0, %1, off"
                     :: "v"(lds_off), "v"(ga) : "memory");
        asm volatile("s_wait_asynccnt 0x0" ::: "memory");
        __syncthreads();                      // panel visible to all waves

        #pragma unroll 4
        for (int k = 0; k < KBLK; k += 4) {
            v2f a;
            a.x = As[l16][k + 2 * hw];
            a.y = As[l16][k + 2 * hw + 1];
            const float* bp = Bm + (size_t)(k0 + k + 2 * hw) * DDIM + n0w + l16;
            v2f b0v, b1v, b2v, b3v;
            b0v.x = bp[0];  b0v.y = bp[DDIM];
            b1v.x = bp[16]; b1v.y = bp[DDIM + 16];
            b2v.x = bp[32]; b2v.y = bp[DDIM + 32];
            b3v.x = bp[48]; b3v.y = bp[DDIM + 48];
            c0 = __builtin_amdgcn_wmma_f32_16x16x4_f32(false, a, false, b0v, (short)0, c0, false, false);
            c1 = __builtin_amdgcn_wmma_f32_16x16x4_f32(false, a, false, b1v, (short)0, c1, false, false);
            c2 = __builtin_amdgcn_wmma_f32_16x16x4_f32(false, a, false, b2v, (short)0, c2, false, false);
            c3 = __builtin_amdgcn_wmma_f32_16x16x4_f32(false, a, false, b3v, (short)0, c3, false, false);
        }
    }

    #pragma unroll
    for (int r = 0; r < 8; ++r) {
        int row = m0 + r + hw * 8;
        #pragma unroll
        for (int j = 0; j < 4; ++j) {
            int col = n0w + j * 16 + l16;
            int idx = row * DDIM + col;
            float acc = (j == 0) ? c0[r] : (j == 1) ? c1[r] : (j == 2) ? c2[r] : c3[r];
            float o;
            if (mode == 0)      o = acc + C2[col];
            else if (mode == 1) o = tanhf(acc + C2[idx]) - A[idx];
            else                o = tanhf(acc + C2[idx]);
            Out[idx] = o;
        }
    }
}

// ---------------------------------------------------------------------------
// Elementwise / reduction helper kernels (Broyden bookkeeping)
// ---------------------------------------------------------------------------
__global__ void k_zero(float* __restrict__ p, size_t n)
{
    size_t i = (size_t)blockIdx.x * blockDim.x + threadIdx.x;
    size_t stride = (size_t)gridDim.x * blockDim.x;
    for (; i < n; i += stride) p[i] = 0.0f;
}

__global__ void k_copy_init(const float* __restrict__ x0,
                            const float* __restrict__ gx,
                            float* __restrict__ x,
                            float* __restrict__ lx,
                            float* __restrict__ upd)
{
    int i = blockIdx.x * blockDim.x + threadIdx.x;
    float v = x0[i];
    x[i] = v;
    lx[i] = v;
    upd[i] = gx[i];
}

__global__ void k_xnew(const float* __restrict__ x,
                       const float* __restrict__ upd,
                       float* __restrict__ xn,
                       const int* __restrict__ act)
{
    if (*act == 0) return;
    int i = blockIdx.x * blockDim.x + threadIdx.x;
    xn[i] = x[i] + upd[i];
}

// partial[256] <- per-block sum of squares (256 blocks x 2048 elems)
__global__ void k_sumsq(const float* __restrict__ v,
                        float* __restrict__ partial,
                        const int* __restrict__ act)
{
    if (act && *act == 0) return;
    __shared__ float s[NT];
    float acc = 0.0f;
    int base = blockIdx.x * 2048 + threadIdx.x;
    #pragma unroll
    for (int j = 0; j < 8; ++j) { float x = v[base + j * 256]; acc += x * x; }
    s[threadIdx.x] = acc;
    __syncthreads();
    for (int o = NT / 2; o > 0; o >>= 1) {
        if (threadIdx.x < o) s[threadIdx.x] += s[threadIdx.x + o];
        __syncthreads();
    }
    if (threadIdx.x == 0) partial[blockIdx.x] = s[0];
}

// sc[0]=low, sc[1]=protect ; act[0] = loop-entry condition
__global__ void k_ctrl_init(const float* __restrict__ partial,
                            float* __restrict__ sc,
                            int* __restrict__ act)
{
    __shared__ float s[NT];
    s[threadIdx.x] = partial[threadIdx.x];
    __syncthreads();
    for (int o = NT / 2; o > 0; o >>= 1) {
        if (threadIdx.x < o) s[threadIdx.x] += s[threadIdx.x + o];
        __syncthreads();
    }
    if (threadIdx.x == 0) {
        float obj = sqrtf(s[0]);          // init_obj
        sc[0] = obj;                      // low
        sc[1] = obj * 1e6f;               // protect = init_obj * (1e6 * E), E=1
        act[0] = (obj >= 1e-8f && obj <= sc[1]) ? 1 : 0;
    }
}

__global__ void k_ctrl_step(const float* __restrict__ partial,
                            float* __restrict__ sc,
                            int* __restrict__ act,
                            int* __restrict__ bet,
                            int nstep)
{
    __shared__ float s[NT];
    s[threadIdx.x] = partial[threadIdx.x];
    __syncthreads();
    for (int o = NT / 2; o > 0; o >>= 1) {
        if (threadIdx.x < o) s[threadIdx.x] += s[threadIdx.x + o];
        __syncthreads();
    }
    if (threadIdx.x == 0) {
        if (act[nstep - 1] == 0) { act[nstep] = 0; bet[nstep] = 0; return; }
        float objn = sqrtf(s[0]);
        int better = objn < sc[0] ? 1 : 0;
        if (better) sc[0] = objn;
        bet[nstep] = better;
        act[nstep] = (objn >= 1e-8f && nstep < TMAX && objn <= sc[1]) ? 1 : 0;
    }
}

// dg = gx_new - gx ; lx = x_new when this step improved the objective
__global__ void k_dg_lx(const float* __restrict__ gn,
                        const float* __restrict__ g,
                        const float* __restrict__ xn,
                        float* __restrict__ dg,
                        float* __restrict__ lx,
                        const int* __restrict__ act,
                        const int* __restrict__ bet)
{
    if (*act == 0) return;
    int i = blockIdx.x * blockDim.x + threadIdx.x;
    dg[i] = gn[i] - g[i];
    if (*bet) lx[i] = xn[i];
}

// out[b*T+t] = dot_d( vec[b,:], planes[t][b,:] )  (xTU / VTx / VTx2)
__global__ void k_dot_bt(const float* __restrict__ vec,
                         const float* __restrict__ planes,
                         float* __restrict__ out,
                         const int* __restrict__ act)
{
    if (*act == 0) return;
    int b = blockIdx.x / TMAX;
    int t = blockIdx.x % TMAX;
    const float* va = vec + (size_t)b * DDIM;
    const float* pa = planes + (size_t)t * BD + (size_t)b * DDIM;
    __shared__ float s[NT];
    float acc = 0.0f;
    for (int j = threadIdx.x; j < DDIM; j += NT) acc += va[j] * pa[j];
    s[threadIdx.x] = acc;
    __syncthreads();
    for (int o = NT / 2; o > 0; o >>= 1) {
        if (threadIdx.x < o) s[threadIdx.x] += s[threadIdx.x + o];
        __syncthreads();
    }
    if (threadIdx.x == 0) out[b * TMAX + t] = s[0];
}

// out[b] = dot_d( a[b,:], c[b,:] )   (denominator)
__global__ void k_dot_b(const float* __restrict__ a,
                        const float* __restrict__ c,
                        float* __restrict__ out,
                        const int* __restrict__ act)
{
    if (*act == 0) return;
    int b = blockIdx.x;
    const float* pa = a + (size_t)b * DDIM;
    const float* pc = c + (size_t)b * DDIM;
    __shared__ float s[NT];
    float acc = 0.0f;
    for (int j = threadIdx.x; j < DDIM; j += NT) acc += pa[j] * pc[j];
    s[threadIdx.x] = acc;
    __syncthreads();
    for (int o = NT / 2; o > 0; o >>= 1) {
        if (threadIdx.x < o) s[threadIdx.x] += s[threadIdx.x + o];
        __syncthreads();
    }
    if (threadIdx.x == 0) out[b] = s[0];
}

// vT = rmatvec(Us, VTs, dx) = -dx + sum_t xTU[b,t] * VTs[t][b,:]
__global__ void k_vt(const float* __restrict__ dx,
                     const float* __restrict__ VTs,
                     const float* __restrict__ xTU,
                     float* __restrict__ vT,
                     const int* __restrict__ act)
{
    if (*act == 0) return;
    int i = blockIdx.x * blockDim.x + threadIdx.x;
    int b = i >> 11;
    float s = -dx[i];
    #pragma unroll
    for (int t = 0; t < TMAX; ++t) s += xTU[b * TMAX + t] * VTs[(size_t)t * BD + i];
    vT[i] = s;
}

// u = (dx - matvec(Us,VTs,dg)) / denom ; NaN->0 ; commit slot planes
__global__ void k_u(const float* __restrict__ dx,
                    const float* __restrict__ dg,
                    const float* __restrict__ VTx,
                    const float* __restrict__ den,
                    const float* __restrict__ vT,
                    float* __restrict__ Us,
                    float* __restrict__ VTs,
                    int slot,
                    const int* __restrict__ act)
{
    if (*act == 0) return;
    int i = blockIdx.x * blockDim.x + threadIdx.x;
    int b = i >> 11;
    float mv = -dg[i];
    #pragma unroll
    for (int t = 0; t < TMAX; ++t) mv += Us[(size_t)t * BD + i] * VTx[b * TMAX + t];
    float u = (dx[i] - mv) / den[b];
    u = (u == u) ? u : 0.0f;                 // jnp.where(isnan(u), 0, u)
    float vv = vT[i];
    vv = (vv == vv) ? vv : 0.0f;             // jnp.where(isnan(vT), 0, vT)
    Us[(size_t)slot * BD + i] = u;
    VTs[(size_t)slot * BD + i] = vv;
}

// upd = -matvec(Us,VTs,gx_new) = gx_new - sum_t Us[t]*VTx2[b,t]
__global__ void k_upd(const float* __restrict__ gn,
                      const float* __restrict__ Us,
                      const float* __restrict__ VTx2,
                      float* __restrict__ upd,
                      const int* __restrict__ act)
{
    if (*act == 0) return;
    int i = blockIdx.x * blockDim.x + threadIdx.x;
    int b = i >> 11;
    float s = gn[i];
    #pragma unroll
    for (int t = 0; t < TMAX; ++t) s -= Us[(size_t)t * BD + i] * VTx2[b * TMAX + t];
    upd[i] = s;
}

// ---------------------------------------------------------------------------
// Host orchestration (graph-capture safe: only kernel launches on `stream`)
// ---------------------------------------------------------------------------
extern "C" void kernel_launch(void* const* d_in, const int* in_sizes, int n_in,
                              void* d_out, int out_size, void* d_ws, size_t ws_size,
                              hipStream_t stream)
{
    const float* x    = (const float*)d_in[0];   // [256,2048]
    const float* x0   = (const float*)d_in[1];   // initial_point
    const float* W    = (const float*)d_in[2];   // [2048,2048]
    const float* U    = (const float*)d_in[3];   // [2048,2048]
    const float* bias = (const float*)d_in[4];   // [2048]
    float* out = (float*)d_out;

    float* w   = (float*)d_ws;                   // needs ~69 MB
    float* xUb = w + 0 * (size_t)BD;
    float* xA  = w + 1 * (size_t)BD;             // x (ping)
    float* xB  = w + 2 * (size_t)BD;             // x_new (pong)
    float* gA  = w + 3 * (size_t)BD;             // gx (ping)
    float* gB  = w + 4 * (size_t)BD;             // gx_new (pong)
    float* upd = w + 5 * (size_t)BD;
    float* lx  = w + 6 * (size_t)BD;
    float* dg  = w + 7 * (size_t)BD;
    float* vT  = w + 8 * (size_t)BD;
    float* Us  = w + 9 * (size_t)BD;             // 12 planes
    float* VTs = w + 21 * (size_t)BD;            // 12 planes
    float* part = w + 33 * (size_t)BD;           // [256]
    float* xTU  = part + 256;                    // [256*12]
    float* VTx  = xTU + BDIM * TMAX;
    float* VTx2 = VTx + BDIM * TMAX;
    float* den  = VTx2 + BDIM * TMAX;            // [256]
    float* sc   = den + 256;                     // low, protect
    int*   act  = (int*)(sc + 8);                // active[13]
    int*   bet  = act + 16;                      // better[13]

    dim3 blk(NT);
    dim3 gGemm(64);         // 16 mtiles x 4 nblocks; 8 waves x 4 tiles each
    dim3 gEW(BD / NT);      // elementwise over B*D
    dim3 gBT(BDIM * TMAX);  // batched rank dots
    dim3 gB1(BDIM);

    // xUb = x @ U + bias   (loop invariant)
    k_gemm_wmma<<<gGemm, blk, 0, stream>>>(x, U, bias, xUb, 0, nullptr);
    // gx0 = tanh(x0 @ W + xUb) - x0
    k_gemm_wmma<<<gGemm, blk, 0, stream>>>(x0, W, xUb, gA, 1, nullptr);
    k_zero<<<dim3(4096), blk, 0, stream>>>(Us, (size_t)24 * BD);  // Us + VTs
    k_copy_init<<<gEW, blk, 0, stream>>>(x0, gA, xA, lx, upd);
    k_sumsq<<<dim3(256), blk, 0, stream>>>(gA, part, nullptr);
    k_ctrl_init<<<dim3(1), blk, 0, stream>>>(part, sc, act);

    float* xc = xA; float* xn = xB;
    float* gc = gA; float* gn = gB;

    for (int i = 1; i <= TMAX; ++i) {
        const int* ai = act + (i - 1);
        k_xnew<<<gEW, blk, 0, stream>>>(xc, upd, xn, ai);
        k_gemm_wmma<<<gGemm, blk, 0, stream>>>(xn, W, xUb, gn, 1, ai);
        k_sumsq<<<dim3(256), blk, 0, stream>>>(gn, part, ai);
        k_ctrl_step<<<dim3(1), blk, 0, stream>>>(part, sc, act, bet, i);
        k_dg_lx<<<gEW, blk, 0, stream>>>(gn, gc, xn, dg, lx, ai, bet + i);
        k_dot_bt<<<gBT, blk, 0, stream>>>(upd, Us, xTU, ai);   // xTU = dx . Us
        k_dot_bt<<<gBT, blk, 0, stream>>>(dg, VTs, VTx, ai);   // VTx = VTs . dg
        k_vt<<<gEW, blk, 0, stream>>>(upd, VTs, xTU, vT, ai);
        k_dot_b<<<gB1, blk, 0, stream>>>(vT, dg, den, ai);     // denom (raw vT)
        k_u<<<gEW, blk, 0, stream>>>(upd, dg, VTx, den, vT, Us, VTs, i - 1, ai);
        k_dot_bt<<<gBT, blk, 0, stream>>>(gn, VTs, VTx2, ai);  // new VTs . gx_new
        k_upd<<<gEW, blk, 0, stream>>>(gn, Us, VTx2, upd, ai);
        float* t1 = xc; xc = xn; xn = t1;
        float* t2 = gc; gc = gn; gn = t2;
    }

    // z = tanh(lx @ W + xUb)
    k_gemm_wmma<<<gGemm, blk, 0, stream>>>(lx, W, xUb, out, 2, nullptr);
}